// GCLMemory_29772713296515
// MI455X (gfx1250) — compile-verified
//
#include <hip/hip_runtime.h>
#include <hip/hip_bf16.h>
#include <math.h>

typedef __attribute__((ext_vector_type(2))) float v2f;
typedef __attribute__((ext_vector_type(8))) float v8f;

#define NDIM 2048
#define KDIM 256
#define WAVES 8
#define EPSV 1e-8f

// ---------------------------------------------------------------------------
// Kernel 1: cos[b,n] = (keys[b,n,:]·q[b,:]) / (max(|keys row|,eps)*max(|q|,eps))
// One wave per 16-row tile; dots via V_WMMA_F32_16X16X4_F32 (exact fp32),
// row sum-of-squares on the VALU in parallel.
// ---------------------------------------------------------------------------
__global__ __launch_bounds__(256) void gcl_score_wmma(
    const float* __restrict__ q,       // [B,K]
    const float* __restrict__ keys,    // [B,N,K]
    float* __restrict__ cosout)        // [B,N]
{
    const int b    = blockIdx.y;
    const int tid  = threadIdx.x;
    const int lane = tid & 31;
    const int wave = tid >> 5;

    __shared__ float qs[KDIM];
    __shared__ float qn_s;
    __shared__ float dotbuf[WAVES][16];
    __shared__ float ssbuf[WAVES][16];

    // stage query row in LDS (coalesced: 256 threads, 256 floats)
    qs[tid] = q[b * KDIM + tid];
    __syncthreads();

    // wave 0 computes query norm once
    if (wave == 0) {
        float ss = 0.f;
        #pragma unroll
        for (int i = 0; i < KDIM / 32; ++i) {
            float v = qs[lane + i * 32];
            ss = fmaf(v, v, ss);
        }
        #pragma unroll
        for (int off = 16; off > 0; off >>= 1) ss += __shfl_xor(ss, off, 32);
        if (lane == 0) qn_s = fmaxf(sqrtf(ss), EPSV);
    }
    __syncthreads();
    const float qn = qn_s;

    // A-fragment addressing per ISA 16x4 f32 layout:
    //   lanes 0-15 : row (n0+lane),    K = {kc, kc+1}
    //   lanes 16-31: row (n0+lane-16), K = {kc+2, kc+3}
    const int n0    = (blockIdx.x * WAVES + wave) * 16;
    const int mrow  = lane & 15;
    const int khalf = (lane >> 4) << 1;
    const float* aptr = keys + ((size_t)b * NDIM + (n0 + mrow)) * KDIM + khalf;

    v8f acc = {0.f, 0.f, 0.f, 0.f, 0.f, 0.f, 0.f, 0.f};
    float ssq = 0.f;

    #pragma unroll 4
    for (int kc = 0; kc < KDIM; kc += 4) {
        v2f a = *(const v2f*)(aptr + kc);
        // B-fragment: query chunk broadcast into every column (N-striping moot)
        v2f bq;
        bq.x = qs[kc + khalf];
        bq.y = qs[kc + khalf + 1];
        acc = __builtin_amdgcn_wmma_f32_16x16x4_f32(
            /*neg_a=*/false, a, /*neg_b=*/false, bq,
            /*c_mod=*/(short)0, acc, /*reuse_a=*/false, /*reuse_b=*/false);
        // row sum-of-squares on the VALU alongside the matrix pipe
        ssq = fmaf(a.x, a.x, ssq);
        ssq = fmaf(a.y, a.y, ssq);
    }
    // combine K-halves: lane L and lane L^16 hold partials of the same row
    ssq += __shfl_xor(ssq, 16, 32);

    // D layout (16x16 f32): lane 0 VGPR r -> D[M=r, N=0]; lane 16 -> D[M=8+r, N=0]
    if ((lane & 15) == 0) {
        const int base = (lane >> 4) * 8;
        #pragma unroll
        for (int r = 0; r < 8; ++r) dotbuf[wave][base + r] = acc[r];
    }
    if (lane < 16) ssbuf[wave][lane] = ssq;
    __syncthreads();

    if (lane < 16) {
        const float kn = fmaxf(sqrtf(ssbuf[wave][lane]), EPSV);
        cosout[(size_t)b * NDIM + n0 + lane] = dotbuf[wave][lane] / (kn * qn);
    }
}

// ---------------------------------------------------------------------------
// Kernel 2: per batch row — softmax(beta*cos), exact top-C mask (1.0 / 1e-16),
// renormalize, pow(gamma), renormalize. Softmax denominator cancels after the
// mask renormalization, so only exp(x - max) is needed.
// One 256-thread block per batch; element i owned by thread (i & 255).
// ---------------------------------------------------------------------------
__global__ __launch_bounds__(256) void gcl_topk_sharpen(
    const float* __restrict__ cosv,    // [B,N]
    const float* __restrict__ beta,    // [B]
    const float* __restrict__ gamma,   // [B]
    const int*   __restrict__ cand,    // [1]
    float* __restrict__ out)           // [B,N]
{
    const int b   = blockIdx.x;
    const int tid = threadIdx.x;

    __shared__ float e[NDIM];
    __shared__ float t[NDIM];
    __shared__ float mk[NDIM];
    __shared__ float redf[256];
    __shared__ int   redi[256];

    const float bet = beta[b];
    const int   C   = cand[0];

    float loc[8];
    #pragma unroll
    for (int j = 0; j < 8; ++j)
        loc[j] = bet * cosv[(size_t)b * NDIM + j * 256 + tid];

    // block max
    float lm = loc[0];
    #pragma unroll
    for (int j = 1; j < 8; ++j) lm = fmaxf(lm, loc[j]);
    redf[tid] = lm;
    __syncthreads();
    for (int off = 128; off > 0; off >>= 1) {
        if (tid < off) redf[tid] = fmaxf(redf[tid], redf[tid + off]);
        __syncthreads();
    }
    const float mx = redf[0];
    __syncthreads();

    #pragma unroll
    for (int j = 0; j < 8; ++j) {
        const int i = j * 256 + tid;
        const float ev = __expf(loc[j] - mx);
        e[i] = ev; t[i] = ev; mk[i] = 1e-16f;
    }
    __syncthreads();

    // exact top-C: C rounds of (max value, min index) block reductions
    for (int c = 0; c < C; ++c) {
        float vm = t[tid];
        #pragma unroll
        for (int j = 1; j < 8; ++j) vm = fmaxf(vm, t[j * 256 + tid]);
        redf[tid] = vm;
        __syncthreads();
        for (int off = 128; off > 0; off >>= 1) {
            if (tid < off) redf[tid] = fmaxf(redf[tid], redf[tid + off]);
            __syncthreads();
        }
        const float maxv = redf[0];

        int li = 1 << 30;
        #pragma unroll
        for (int j = 0; j < 8; ++j) {
            const int i = j * 256 + tid;
            if (t[i] == maxv) li = min(li, i);
        }
        redi[tid] = li;
        __syncthreads();
        for (int off = 128; off > 0; off >>= 1) {
            if (tid < off) redi[tid] = min(redi[tid], redi[tid + off]);
            __syncthreads();
        }
        const int idx = redi[0];
        __syncthreads();
        if (idx < NDIM && (idx & 255) == tid) { mk[idx] = 1.0f; t[idx] = -1.0f; }
        __syncthreads();
    }

    // sum(e*mask)
    float s1 = 0.f;
    #pragma unroll
    for (int j = 0; j < 8; ++j) {
        const int i = j * 256 + tid;
        s1 += e[i] * mk[i];
    }
    redf[tid] = s1;
    __syncthreads();
    for (int off = 128; off > 0; off >>= 1) {
        if (tid < off) redf[tid] += redf[tid + off];
        __syncthreads();
    }
    const float S1 = redf[0];
    __syncthreads();

    // sharpen: w = (e*mk/S1)^gamma, then normalize
    const float g = gamma[b];
    float w[8];
    float s2 = 0.f;
    #pragma unroll
    for (int j = 0; j < 8; ++j) {
        const int i = j * 256 + tid;
        const float wn = e[i] * mk[i] / S1;
        w[j] = __powf(wn, g);
        s2 += w[j];
    }
    redf[tid] = s2;
    __syncthreads();
    for (int off = 128; off > 0; off >>= 1) {
        if (tid < off) redf[tid] += redf[tid + off];
        __syncthreads();
    }
    const float S2 = redf[0];

    #pragma unroll
    for (int j = 0; j < 8; ++j)
        out[(size_t)b * NDIM + j * 256 + tid] = w[j] / S2;
}

extern "C" void kernel_launch(void* const* d_in, const int* in_sizes, int n_in,
                              void* d_out, int out_size, void* d_ws, size_t ws_size,
                              hipStream_t stream) {
    const float* q     = (const float*)d_in[0];   // [B,K]
    const float* beta  = (const float*)d_in[1];   // [B]
    const float* gamma = (const float*)d_in[2];   // [B]
    const float* keys  = (const float*)d_in[3];   // [B,N,K]
    const int*   cand  = (const int*)d_in[4];     // [1] (=32)
    float* out = (float*)d_out;

    const int B = in_sizes[1];                    // 128
    float* cosbuf = (float*)d_ws;                 // B*N floats = 1 MB scratch

    dim3 g1(NDIM / (16 * WAVES), B);              // (16, 128) blocks, 1 wave/tile
    gcl_score_wmma<<<g1, 256, 0, stream>>>(q, keys, cosbuf);
    gcl_topk_sharpen<<<B, 256, 0, stream>>>(cosbuf, beta, gamma, cand, out);
}